// GraphPolicyNetwork_6416681140225
// MI455X (gfx1250) — compile-verified
//
#include <hip/hip_runtime.h>
#include <hip/hip_bf16.h>

#define N_NODES 50000
#define N_EDGES 1600000
#define EN_TOT  (N_EDGES + N_NODES)
#define IN_DIM  32
#define HID     128
#define NLAYER  3
#define PI_F    3.14159265358979323846f

typedef float v2f __attribute__((ext_vector_type(2)));
typedef float v8f __attribute__((ext_vector_type(8)));
typedef int   v4i __attribute__((ext_vector_type(4)));

#if __has_builtin(__builtin_amdgcn_global_load_async_to_lds_b128)
#define HAVE_ASYNC_LDS 1
#else
#define HAVE_ASYNC_LDS 0
#endif

__device__ __forceinline__ void async_wait0() {
#if __has_builtin(__builtin_amdgcn_s_wait_asynccnt)
  __builtin_amdgcn_s_wait_asynccnt(0);
#else
  asm volatile("s_wait_asynccnt 0x0" ::: "memory");
#endif
}

// ---------------- utility ----------------
__global__ __launch_bounds__(256) void fill_k(float* __restrict__ p, float v, int n) {
  int i = blockIdx.x * blockDim.x + threadIdx.x;
  if (i < n) p[i] = v;
}

__device__ __forceinline__ void atomicMaxF(float* addr, float val) {
  int* ai = (int*)addr;
  int old = __float_as_int(*addr);
  while (__int_as_float(old) < val) {
    int prev = atomicCAS(ai, old, __float_as_int(val));
    if (prev == old) break;
    old = prev;
  }
}

__device__ __forceinline__ float sigmoidf_(float x) { return 1.0f / (1.0f + expf(-x)); }

// ---- stage a 16xK fp32 A-tile into LDS (padded stride K+4), async if available ----
template<int K>
__device__ __forceinline__ void stage_A(const float* __restrict__ A, int m0, float* As) {
  constexpr int K4 = K / 4;
  for (int i = threadIdx.x; i < 16 * K4; i += 256) {
    const int r  = i / K4;
    const int cv = (i - r * K4) * 4;
    float* l = &As[r * (K + 4) + cv];
    const float* g = A + (size_t)(m0 + r) * K + cv;
#if HAVE_ASYNC_LDS
    __builtin_amdgcn_global_load_async_to_lds_b128((v4i*)g, (v4i*)l, 0, 0);
#else
    *(float4*)l = *(const float4*)g;
#endif
  }
#if HAVE_ASYNC_LDS
  async_wait0();
#endif
  __syncthreads();
}

// ---------------- WMMA fp32 GEMM: out[n,128] = act(A[n,K] @ W[K,128] + bias) ----------------
// Block = 256 thr = 8 waves; A tile staged in LDS once per block; each wave owns a
// 16x16 C tile via V_WMMA_F32_16X16X4_F32 (K/4 steps).
template<int K, bool RELU>
__global__ __launch_bounds__(256) void wmma_gemm_k(const float* __restrict__ A,
                                                   const float* __restrict__ W,
                                                   const float* __restrict__ bias,
                                                   float* __restrict__ out)
{
  __shared__ float As[16 * (K + 4)];
  const int m0 = blockIdx.x << 4;
  stage_A<K>(A, m0, As);

  const int lane = threadIdx.x & 31;
  const int wave = threadIdx.x >> 5;
  const int n0   = wave << 4;
  const int row  = lane & 15;
  const int half = lane >> 4;
  const float* Arow = &As[row * (K + 4)];
  v8f c = {};
  for (int k0 = 0; k0 < K; k0 += 4) {
    const int ka = k0 + 2 * half;
    v2f a, b;
    a.x = Arow[ka];
    a.y = Arow[ka + 1];
    b.x = W[(size_t)ka * HID + n0 + row];
    b.y = W[(size_t)(ka + 1) * HID + n0 + row];
    c = __builtin_amdgcn_wmma_f32_16x16x4_f32(false, a, false, b, (short)0, c, false, false);
  }
  const float bv = bias[n0 + row];
  #pragma unroll
  for (int j = 0; j < 8; ++j) {
    float v = c[j] + bv;
    if (RELU) v = fmaxf(v, 0.0f);
    out[(size_t)(m0 + j + 8 * half) * HID + n0 + row] = v;
  }
}

// ---------------- fused dual GEMM (shared A): hs = A@W0+b0, hd = A@W1+b1 ----------------
__global__ __launch_bounds__(256) void wmma_gemm2_k(const float* __restrict__ A,
                                                    const float* __restrict__ W0,
                                                    const float* __restrict__ b0,
                                                    float* __restrict__ o0,
                                                    const float* __restrict__ W1,
                                                    const float* __restrict__ b1,
                                                    float* __restrict__ o1)
{
  constexpr int K = HID;
  __shared__ float As[16 * (K + 4)];
  const int m0 = blockIdx.x << 4;
  stage_A<K>(A, m0, As);

  const int lane = threadIdx.x & 31;
  const int wave = threadIdx.x >> 5;
  const int n0   = wave << 4;
  const int row  = lane & 15;
  const int half = lane >> 4;
  const float* Arow = &As[row * (K + 4)];
  v8f c0 = {}, c1 = {};
  for (int k0 = 0; k0 < K; k0 += 4) {
    const int ka = k0 + 2 * half;
    v2f a, b;
    a.x = Arow[ka];
    a.y = Arow[ka + 1];
    b.x = W0[(size_t)ka * HID + n0 + row];
    b.y = W0[(size_t)(ka + 1) * HID + n0 + row];
    c0 = __builtin_amdgcn_wmma_f32_16x16x4_f32(false, a, false, b, (short)0, c0, false, false);
    b.x = W1[(size_t)ka * HID + n0 + row];
    b.y = W1[(size_t)(ka + 1) * HID + n0 + row];
    c1 = __builtin_amdgcn_wmma_f32_16x16x4_f32(false, a, false, b, (short)0, c1, false, false);
  }
  const float bv0 = b0[n0 + row];
  const float bv1 = b1[n0 + row];
  #pragma unroll
  for (int j = 0; j < 8; ++j) {
    const size_t idx = (size_t)(m0 + j + 8 * half) * HID + n0 + row;
    o0[idx] = c0[j] + bv0;
    o1[idx] = c1[j] + bv1;
  }
}

// ---------------- edge pass 1: scores + segment max ----------------
// one wave32 per edge; lane l owns feats [4l..4l+3] (head = l/8); attn flat [H*O]=128.
__global__ __launch_bounds__(256) void edge_score_k(const float* __restrict__ hs,
                                                    const float* __restrict__ hd,
                                                    const int* __restrict__ src,
                                                    const int* __restrict__ dst,
                                                    const float* __restrict__ attn,
                                                    float* __restrict__ sc,
                                                    float* __restrict__ m)
{
  const int e = (blockIdx.x * blockDim.x + threadIdx.x) >> 5;
  if (e >= EN_TOT) return;
  const int lane = threadIdx.x & 31;
  const int u = (e < N_EDGES) ? src[e] : (e - N_EDGES);
  const int v = (e < N_EDGES) ? dst[e] : (e - N_EDGES);
  const float4 a4 = *(const float4*)(hs + (size_t)u * HID + 4 * lane);
  const float4 b4 = *(const float4*)(hd + (size_t)v * HID + 4 * lane);
  const float4 w4 = *(const float4*)(attn + 4 * lane);
  float p = 0.0f;
  {
    float x;
    x = a4.x + b4.x; p += w4.x * (x > 0.0f ? x : 0.2f * x);
    x = a4.y + b4.y; p += w4.y * (x > 0.0f ? x : 0.2f * x);
    x = a4.z + b4.z; p += w4.z * (x > 0.0f ? x : 0.2f * x);
    x = a4.w + b4.w; p += w4.w * (x > 0.0f ? x : 0.2f * x);
  }
  // reduce within each 8-lane head group
  p += __shfl_xor(p, 1, 8);
  p += __shfl_xor(p, 2, 8);
  p += __shfl_xor(p, 4, 8);
  if ((lane & 7) == 0) {
    const int h = lane >> 3;
    sc[(size_t)e * 4 + h] = p;
    atomicMaxF(&m[(size_t)v * 4 + h], p);
  }
}

// ---------------- edge pass 2: exp + segment sum ----------------
__global__ __launch_bounds__(256) void edge_exp_k(float* __restrict__ sc,
                                                  const int* __restrict__ dst,
                                                  const float* __restrict__ m,
                                                  float* __restrict__ s)
{
  const int i = blockIdx.x * blockDim.x + threadIdx.x;
  if (i >= EN_TOT * 4) return;
  const int e = i >> 2, h = i & 3;
  const int v = (e < N_EDGES) ? dst[e] : (e - N_EDGES);
  const float ex = expf(sc[i] - m[(size_t)v * 4 + h]);
  sc[i] = ex;
  atomicAdd(&s[(size_t)v * 4 + h], ex);
}

// ---------------- edge pass 3: normalized weighted aggregation ----------------
__global__ __launch_bounds__(256) void edge_agg_k(const float* __restrict__ hs,
                                                  const float* __restrict__ sc,
                                                  const float* __restrict__ s,
                                                  const int* __restrict__ src,
                                                  const int* __restrict__ dst,
                                                  float* __restrict__ acc)
{
  const int e = (blockIdx.x * blockDim.x + threadIdx.x) >> 5;
  if (e >= EN_TOT) return;
  const int lane = threadIdx.x & 31;
  const int u = (e < N_EDGES) ? src[e] : (e - N_EDGES);
  const int v = (e < N_EDGES) ? dst[e] : (e - N_EDGES);
  const int h = lane >> 3;
  const float alpha = sc[(size_t)e * 4 + h] / (s[(size_t)v * 4 + h] + 1e-9f);
  const float4 a4 = *(const float4*)(hs + (size_t)u * HID + 4 * lane);
  float* d = acc + (size_t)v * HID + 4 * lane;
  atomicAdd(d + 0, alpha * a4.x);
  atomicAdd(d + 1, alpha * a4.y);
  atomicAdd(d + 2, alpha * a4.z);
  atomicAdd(d + 3, alpha * a4.w);
}

// ---------------- LayerNorm + ReLU, in-place on h ----------------
// x = 2h + acc ; ln ; relu -> h.  One wave per node, lane owns 4 elems.
__global__ __launch_bounds__(256) void node_norm_k(float* __restrict__ h,
                                                   const float* __restrict__ acc,
                                                   const float* __restrict__ g,
                                                   const float* __restrict__ b)
{
  const int n = (blockIdx.x * blockDim.x + threadIdx.x) >> 5;
  if (n >= N_NODES) return;
  const int lane = threadIdx.x & 31;
  float4 hv = *(const float4*)(h + (size_t)n * HID + 4 * lane);
  const float4 av = *(const float4*)(acc + (size_t)n * HID + 4 * lane);
  float x0 = 2.0f * hv.x + av.x, x1 = 2.0f * hv.y + av.y;
  float x2 = 2.0f * hv.z + av.z, x3 = 2.0f * hv.w + av.w;
  float sum = x0 + x1 + x2 + x3;
  float sq  = x0 * x0 + x1 * x1 + x2 * x2 + x3 * x3;
  #pragma unroll
  for (int msk = 16; msk >= 1; msk >>= 1) {
    sum += __shfl_xor(sum, msk, 32);
    sq  += __shfl_xor(sq,  msk, 32);
  }
  const float mu = sum * (1.0f / HID);
  const float var = sq * (1.0f / HID) - mu * mu;
  const float rstd = rsqrtf(var + 1e-5f);
  const float4 gv = *(const float4*)(g + 4 * lane);
  const float4 bv = *(const float4*)(b + 4 * lane);
  hv.x = fmaxf((x0 - mu) * rstd * gv.x + bv.x, 0.0f);
  hv.y = fmaxf((x1 - mu) * rstd * gv.y + bv.y, 0.0f);
  hv.z = fmaxf((x2 - mu) * rstd * gv.z + bv.z, 0.0f);
  hv.w = fmaxf((x3 - mu) * rstd * gv.w + bv.w, 0.0f);
  *(float4*)(h + (size_t)n * HID + 4 * lane) = hv;
}

// ---------------- graph mean: out[c] += sum_r h[r,c]/N ----------------
__global__ __launch_bounds__(128) void graph_mean_k(const float* __restrict__ h,
                                                    float* __restrict__ out)
{
  const int col = threadIdx.x;  // blockDim = 128
  float a = 0.0f;
  for (int r = blockIdx.x; r < N_NODES; r += gridDim.x) a += h[(size_t)r * HID + col];
  atomicAdd(&out[col], a * (1.0f / (float)N_NODES));
}

// ---------------- heads: logits + spawn params. one wave per node ----------------
__global__ __launch_bounds__(256) void heads_k(const float* __restrict__ z,
                                               const float* __restrict__ Wa, const float* __restrict__ ba,
                                               const float* __restrict__ Wg, const float* __restrict__ bg,
                                               const float* __restrict__ Wal, const float* __restrict__ bal,
                                               const float* __restrict__ Wn, const float* __restrict__ bn,
                                               const float* __restrict__ Wt, const float* __restrict__ bt,
                                               float* __restrict__ logits, float* __restrict__ spawn)
{
  const int n = (blockIdx.x * blockDim.x + threadIdx.x) >> 5;
  if (n >= N_NODES) return;
  const int lane = threadIdx.x & 31;
  const int i0 = 4 * lane;
  const float4 zv = *(const float4*)(z + (size_t)n * HID + i0);
  float p0 = 0, p1 = 0, p2 = 0, p3 = 0, p4 = 0, p5 = 0;
  #pragma unroll
  for (int k = 0; k < 4; ++k) {
    const float zk = (&zv.x)[k];
    const int i = i0 + k;
    p0 += zk * Wa[i * 2 + 0];
    p1 += zk * Wa[i * 2 + 1];
    p2 += zk * Wg[i];
    p3 += zk * Wal[i];
    p4 += zk * Wn[i];
    p5 += zk * Wt[i];
  }
  #pragma unroll
  for (int msk = 16; msk >= 1; msk >>= 1) {
    p0 += __shfl_xor(p0, msk, 32); p1 += __shfl_xor(p1, msk, 32);
    p2 += __shfl_xor(p2, msk, 32); p3 += __shfl_xor(p3, msk, 32);
    p4 += __shfl_xor(p4, msk, 32); p5 += __shfl_xor(p5, msk, 32);
  }
  if (lane == 0) {
    logits[(size_t)n * 2 + 0] = p0 + ba[0];
    logits[(size_t)n * 2 + 1] = p1 + ba[1];
    spawn[(size_t)n * 4 + 0] = 5.0f * sigmoidf_(p2 + bg[0]);
    spawn[(size_t)n * 4 + 1] = 2.0f * sigmoidf_(p3 + bal[0]);
    spawn[(size_t)n * 4 + 2] = 1.0f * sigmoidf_(p4 + bn[0]);
    spawn[(size_t)n * 4 + 3] = tanhf(p5 + bt[0]) * PI_F;
  }
}

// ---------------- host orchestration ----------------
extern "C" void kernel_launch(void* const* d_in, const int* in_sizes, int n_in,
                              void* d_out, int out_size, void* d_ws, size_t ws_size,
                              hipStream_t stream) {
  const float* node_feats = (const float*)d_in[0];
  const int*   src   = (const int*)d_in[1];
  const int*   dst   = (const int*)d_in[2];
  const float* W_in  = (const float*)d_in[3];
  const float* b_in  = (const float*)d_in[4];
  const float* W_src = (const float*)d_in[5];
  const float* b_src = (const float*)d_in[6];
  const float* W_dst = (const float*)d_in[7];
  const float* b_dst = (const float*)d_in[8];
  const float* attn  = (const float*)d_in[9];
  const float* ln_g  = (const float*)d_in[10];
  const float* ln_b  = (const float*)d_in[11];
  const float* W1 = (const float*)d_in[12]; const float* b1 = (const float*)d_in[13];
  const float* W2 = (const float*)d_in[14]; const float* b2 = (const float*)d_in[15];
  const float* Wa = (const float*)d_in[16]; const float* ba = (const float*)d_in[17];
  const float* Wg = (const float*)d_in[18]; const float* bg = (const float*)d_in[19];
  const float* Wal= (const float*)d_in[20]; const float* bal= (const float*)d_in[21];
  const float* Wn = (const float*)d_in[22]; const float* bn = (const float*)d_in[23];
  const float* Wt = (const float*)d_in[24]; const float* bt = (const float*)d_in[25];

  float* out = (float*)d_out;
  float* ws  = (float*)d_ws;

  const size_t NH = (size_t)N_NODES * HID;       // 6,400,000
  float* h   = ws;
  float* hs  = ws + NH;
  float* hd  = ws + 2 * NH;
  float* acc = ws + 3 * NH;
  float* sc  = ws + 4 * NH;                       // (E+N)*4 = 6,600,000
  float* m   = sc + (size_t)EN_TOT * 4;
  float* s   = m + (size_t)N_NODES * 4;
  float* z1  = hs;   // reuse after last GAT layer
  float* z   = hd;   // reuse after last GAT layer

  const int ROWS16 = N_NODES / 16;                // 3125 (exact)
  const int NH4    = N_NODES * 4;
  const int edgeWaveBlocks = (EN_TOT + 7) / 8;    // 8 edge-waves per 256-thr block
  const int ehBlocks = (EN_TOT * 4 + 255) / 256;
  const int nodeWaveBlocks = (N_NODES + 7) / 8;

  // input projection: h = node_feats @ W_in + b_in
  wmma_gemm_k<IN_DIM, false><<<ROWS16, 256, 0, stream>>>(node_feats, W_in, b_in, h);

  for (int l = 0; l < NLAYER; ++l) {
    const float* Ws = W_src + (size_t)l * HID * HID;
    const float* Wd = W_dst + (size_t)l * HID * HID;
    // fused: hs = h@Ws+bs, hd = h@Wd+bd (A staged once)
    wmma_gemm2_k<<<ROWS16, 256, 0, stream>>>(h, Ws, b_src + l * HID, hs,
                                             Wd, b_dst + l * HID, hd);

    fill_k<<<(NH4 + 255) / 256, 256, 0, stream>>>(m, -1e30f, NH4);
    fill_k<<<(NH4 + 255) / 256, 256, 0, stream>>>(s, 0.0f, NH4);
    fill_k<<<(int)((NH + 255) / 256), 256, 0, stream>>>(acc, 0.0f, (int)NH);

    edge_score_k<<<edgeWaveBlocks, 256, 0, stream>>>(hs, hd, src, dst, attn + l * HID, sc, m);
    edge_exp_k<<<ehBlocks, 256, 0, stream>>>(sc, dst, m, s);
    edge_agg_k<<<edgeWaveBlocks, 256, 0, stream>>>(hs, sc, s, src, dst, acc);
    node_norm_k<<<nodeWaveBlocks, 256, 0, stream>>>(h, acc, ln_g + l * HID, ln_b + l * HID);
  }

  // graph_emb = mean over nodes -> out[0:128]
  fill_k<<<1, 256, 0, stream>>>(out, 0.0f, HID);
  graph_mean_k<<<512, 128, 0, stream>>>(h, out);

  // node_emb -> out[128 : 128+N*128]
  (void)hipMemcpyAsync(out + HID, h, NH * sizeof(float), hipMemcpyDeviceToDevice, stream);

  // MLP: z = relu(relu(h@W1+b1)@W2+b2)
  wmma_gemm_k<HID, true><<<ROWS16, 256, 0, stream>>>(h, W1, b1, z1);
  wmma_gemm_k<HID, true><<<ROWS16, 256, 0, stream>>>(z1, W2, b2, z);

  // heads
  float* logits = out + HID + NH;
  float* spawn  = logits + (size_t)N_NODES * 2;
  heads_k<<<nodeWaveBlocks, 256, 0, stream>>>(z, Wa, ba, Wg, bg, Wal, bal, Wn, bn, Wt, bt,
                                              logits, spawn);
}